// DenseGrid_85186381348922
// MI455X (gfx1250) — compile-verified
//
#include <hip/hip_runtime.h>

// DenseGrid trilinear interpolation, MI455X (gfx1250, wave32).
//
// N = 2,097,152 points; codebook = 128^3 x 18 f32 (151 MB, resident in the
// 192 MB global L2); output = N x 18 f32 (151 MB streamed out).
// Pure gather/L2-request-rate problem (~300 FLOPs vs 576 gathered B/point).
//
// Strategy:
//   - x-corner PAIRING: corners (dx=0,dx=1) are contiguous rows idx, idx+1
//     (144 B) -> one gather span per (y,z) pair, every stencil cacheline
//     requested exactly once (~264 L2 requests/wave, irreducible).
//   - codebook gathers: RT temporal policy -> stays resident in L2.
//   - pts loads: non-temporal (streamed once).
//   - OUTPUT VIA LDS + ASYNC DMA: each block stages its 18 KB output tile in
//     LDS (conflict-free b64 stores), then copies it out contiguously with
//     GLOBAL_STORE_ASYNC_FROM_LDS_B128 (ASYNCcnt path): 512 B fully
//     contiguous per wave-instruction = full-line NT writes (18 line
//     transactions/wave instead of ~90 partial-line ones, and no
//     read-modify-write risk from NT early eviction between partial writes).
//   - global_prefetch_b8 on both cachelines of each 144 B gather span.

#define RES  128
#define FEAT 18

typedef float f2 __attribute__((ext_vector_type(2)));

__global__ __launch_bounds__(256) void densegrid_trilerp(
    const float* __restrict__ pts,      // [N,3]
    const float* __restrict__ cb,       // [RES^3, FEAT]
    const float* __restrict__ tr,       // [4,4] row-major
    float* __restrict__ out,            // [N, FEAT]
    int N)
{
  __shared__ __align__(16) float tile[256 * FEAT];   // 18 KB block output tile

  const int tid = threadIdx.x;
  const int p   = blockIdx.x * blockDim.x + tid;
  // Uniform per block: fast path only when the whole block is in range.
  const bool full_block = (blockIdx.x + 1) * blockDim.x <= (unsigned)N;

  f2 acc[9];
  #pragma unroll
  for (int j = 0; j < 9; ++j) acc[j] = (f2){0.0f, 0.0f};

  if (p < N) {
    // ---- inverse of transform[:3,:3] (uniform -> scalar float SALU path)
    const float m00 = tr[0], m01 = tr[1], m02 = tr[2],  t0 = tr[3];
    const float m10 = tr[4], m11 = tr[5], m12 = tr[6],  t1 = tr[7];
    const float m20 = tr[8], m21 = tr[9], m22 = tr[10], t2 = tr[11];
    const float c00 = m11 * m22 - m12 * m21;
    const float c01 = m12 * m20 - m10 * m22;
    const float c02 = m10 * m21 - m11 * m20;
    const float rd  = 1.0f / (m00 * c00 + m01 * c01 + m02 * c02);
    const float i00 = c00 * rd, i01 = (m02 * m21 - m01 * m22) * rd, i02 = (m01 * m12 - m02 * m11) * rd;
    const float i10 = c01 * rd, i11 = (m00 * m22 - m02 * m20) * rd, i12 = (m02 * m10 - m00 * m12) * rd;
    const float i20 = c02 * rd, i21 = (m01 * m20 - m00 * m21) * rd, i22 = (m00 * m11 - m01 * m10) * rd;

    // ---- point load (non-temporal; merges to global_load_b96 th:TH_LOAD_NT)
    const float qx = __builtin_nontemporal_load(pts + 3 * p + 0) - t0;
    const float qy = __builtin_nontemporal_load(pts + 3 * p + 1) - t1;
    const float qz = __builtin_nontemporal_load(pts + 3 * p + 2) - t2;
    const float px = i00 * qx + i01 * qy + i02 * qz;
    const float py = i10 * qx + i11 * qy + i12 * qz;
    const float pz = i20 * qx + i21 * qy + i22 * qz;

    // ---- trilinear weights / clamped lattice indices (reference semantics)
    const float fx = px * (float)(RES - 1);
    const float fy = py * (float)(RES - 1);
    const float fz = pz * (float)(RES - 1);
    const float f0x = floorf(fx), f0y = floorf(fy), f0z = floorf(fz);
    const float wx1 = fx - f0x, wy1 = fy - f0y, wz1 = fz - f0z;
    const float wx0 = 1.0f - wx1, wy0 = 1.0f - wy1, wz0 = 1.0f - wz1;

    int i0x = (int)f0x; i0x = i0x < 0 ? 0 : (i0x > RES - 1 ? RES - 1 : i0x);
    int i0y = (int)f0y; i0y = i0y < 0 ? 0 : (i0y > RES - 1 ? RES - 1 : i0y);
    int i0z = (int)f0z; i0z = i0z < 0 ? 0 : (i0z > RES - 1 ? RES - 1 : i0z);
    const int i1x = i0x + 1 > RES - 1 ? RES - 1 : i0x + 1;
    const int i1y = i0y + 1 > RES - 1 ? RES - 1 : i0y + 1;
    const int i1z = i0z + 1 > RES - 1 ? RES - 1 : i0z + 1;

    // Fold x-clamp into pair weights; rows bx, bx+1 are contiguous 144 B.
    int bx = i0x;
    float wxA = wx0, wxB = wx1;
    if (i1x == i0x) { bx = RES - 2; wxA = 0.0f; wxB = wx0 + wx1; }

    const int y0 = i0y * RES,       y1 = i1y * RES;
    const int z0 = i0z * RES * RES, z1 = i1z * RES * RES;

    int pbase[4];
    pbase[0] = bx + y0 + z0;  pbase[1] = bx + y1 + z0;
    pbase[2] = bx + y0 + z1;  pbase[3] = bx + y1 + z1;

    float wyz[4];
    wyz[0] = wy0 * wz0;  wyz[1] = wy1 * wz0;
    wyz[2] = wy0 * wz1;  wyz[3] = wy1 * wz1;

    // ---- prefetch both cachelines of each 144 B span (global_prefetch_b8)
    #pragma unroll
    for (int c = 0; c < 4; ++c) {
      const char* base = (const char*)(cb + pbase[c] * FEAT);
      __builtin_prefetch(base, 0, 3);
      __builtin_prefetch(base + 140, 0, 3);
    }

    // ---- gather 144 B per (y,z) pair + weighted accumulate (ref order)
    #pragma unroll
    for (int c = 0; c < 4; ++c) {
      const f2* rows = (const f2*)(cb + pbase[c] * FEAT);  // 8 B aligned
      const float wA = wxA * wyz[c];   // row bx   (dx = 0)
      const float wB = wxB * wyz[c];   // row bx+1 (dx = 1)
      #pragma unroll
      for (int j = 0; j < 9; ++j) {
        const f2 rA = rows[j];
        const f2 rB = rows[j + 9];
        acc[j] += wA * rA;
        acc[j] += wB * rB;
      }
    }
  }

  if (full_block) {
    // ---- stage in LDS: dword offset tid*18 + 2j -> 18*tid mod 64 injective
    //      over 32 lanes => conflict-free ds_store_b64
    f2* l = (f2*)&tile[tid * FEAT];
    #pragma unroll
    for (int j = 0; j < 9; ++j) l[j] = acc[j];

    __syncthreads();

    // ---- contiguous copy-out: 256*72 B = 1152 x 16 B chunks.
    //      GLOBAL_STORE_ASYNC_FROM_LDS_B128: per lane 16 B LDS->global,
    //      consecutive lanes -> consecutive chunks -> 512 B/instr, full-line
    //      NT writes. Tracked on ASYNCcnt.
    const float* gbase = out + (size_t)blockIdx.x * (256 * FEAT);
    const unsigned lbase = (unsigned)(uintptr_t)&tile[0];

    #pragma unroll
    for (int k = 0; k < 4; ++k) {
      const int chunk = tid + k * 256;
      const float* gaddr = gbase + chunk * 4;
      const unsigned laddr = lbase + chunk * 16;
      asm volatile("global_store_async_from_lds_b128 %0, %1, off th:TH_STORE_NT"
                   :: "v"(gaddr), "v"(laddr) : "memory");
    }
    if (tid < 128) {                         // remaining 128 chunks
      const int chunk = tid + 1024;
      const float* gaddr = gbase + chunk * 4;
      const unsigned laddr = lbase + chunk * 16;
      asm volatile("global_store_async_from_lds_b128 %0, %1, off th:TH_STORE_NT"
                   :: "v"(gaddr), "v"(laddr) : "memory");
    }
    // LDS is dead after this; S_ENDPGM also implies wait-idle. Explicit wait
    // keeps the wave's async traffic fully retired before exit.
    asm volatile("s_wait_asynccnt 0x0" ::: "memory");
  } else if (p < N) {
    // Ragged tail block (not hit for N = 2,097,152): direct NT stores.
    f2* o = (f2*)(out + p * FEAT);
    #pragma unroll
    for (int j = 0; j < 9; ++j) __builtin_nontemporal_store(acc[j], o + j);
  }
}

extern "C" void kernel_launch(void* const* d_in, const int* in_sizes, int n_in,
                              void* d_out, int out_size, void* d_ws, size_t ws_size,
                              hipStream_t stream) {
  const float* pts = (const float*)d_in[0];   // [4,524288,3] f32
  const float* cb  = (const float*)d_in[1];   // [128^3, 18]  f32
  const float* tr  = (const float*)d_in[2];   // [4,4]        f32
  float* out = (float*)d_out;                 // [N, 18]      f32

  const int N = in_sizes[0] / 3;              // 2,097,152 points
  const int block = 256;                      // 8 wave32 waves
  const int grid  = (N + block - 1) / block;

  hipLaunchKernelGGL(densegrid_trilerp, dim3(grid), dim3(block), 0, stream,
                     pts, cb, tr, out, N);
}